// Type1MoE_6227702579634
// MI455X (gfx1250) — compile-verified
//
#include <hip/hip_runtime.h>
#include <hip/hip_bf16.h>

// Problem constants (from reference): B=4, L=1024, N=8, D=512, E=3
#define DIM      512
#define NEXP     3
#define S_TOK    32768          // B*N*L
#define CAP      10923          // ceil(S/E * 1.0)
#define CAPR     10944          // CAP padded to multiple of 64
#define MTILES   171            // ceil(CAP/64)
#define OUT_ELTS 16777216       // B*L*N*D
#define PITCH    66             // LDS row pitch in floats (bank-conflict-free)

typedef __attribute__((ext_vector_type(2))) float v2f;
typedef __attribute__((ext_vector_type(4))) float v4f;
typedef __attribute__((ext_vector_type(8))) float v8f;

// token s = ((b*N + n)*L + l)  ->  element offset of row in features/out [B,L,N,D]
// = b*(L*N*D) + l*(N*D) + n*D  (identical formula for input and output)
__device__ __forceinline__ long long tok_base(int s) {
    long long b = (s >> 13);          // s / (N*L)
    long long n = (s >> 10) & 7;      // (s / L) % N
    long long l = s & 1023;           // s % L
    return (b << 22) + (l << 12) + (n << 9);
}

// ---------------------------------------------------------------------------
// 0) zero the small atomic accumulators
__global__ void moe_init_kernel(float* sum_gates) {
    if (threadIdx.x < NEXP) sum_gates[threadIdx.x] = 0.0f;
}

// ---------------------------------------------------------------------------
// 1) gating: one wave32 per token. logits = x@wg, softmax, argmax, l_aux sums
__global__ __launch_bounds__(256) void moe_gate_kernel(
    const float* __restrict__ x, const float* __restrict__ wg,
    float* __restrict__ gate_out, int* __restrict__ idx_out,
    float* __restrict__ sum_gates)
{
    __shared__ float wg_s[DIM * NEXP];   // 6 KB
    __shared__ float blk_sum[NEXP];
    const int tid = threadIdx.x;
    for (int i = tid; i < DIM * NEXP; i += 256) wg_s[i] = wg[i];
    if (tid < NEXP) blk_sum[tid] = 0.0f;
    __syncthreads();

    const int wave = tid >> 5, lane = tid & 31;
    const int s = blockIdx.x * 8 + wave;
    const float* xp = x + tok_base(s) + lane * 16;

    float a0 = 0.f, a1 = 0.f, a2 = 0.f;
#pragma unroll
    for (int i = 0; i < 16; i += 4) {
        v4f xv = *(const v4f*)(xp + i);
#pragma unroll
        for (int j = 0; j < 4; ++j) {
            const float xs = xv[j];
            const int d = lane * 16 + i + j;
            a0 = fmaf(xs, wg_s[d * NEXP + 0], a0);
            a1 = fmaf(xs, wg_s[d * NEXP + 1], a1);
            a2 = fmaf(xs, wg_s[d * NEXP + 2], a2);
        }
    }
#pragma unroll
    for (int off = 16; off > 0; off >>= 1) {
        a0 += __shfl_xor(a0, off, 32);
        a1 += __shfl_xor(a1, off, 32);
        a2 += __shfl_xor(a2, off, 32);
    }
    if (lane == 0) {
        const float m  = fmaxf(a0, fmaxf(a1, a2));
        const float e0 = __expf(a0 - m), e1 = __expf(a1 - m), e2 = __expf(a2 - m);
        const float inv = 1.0f / (e0 + e1 + e2);
        const float p0 = e0 * inv, p1 = e1 * inv, p2 = e2 * inv;
        int   am = 0;  float g = p0;
        if (p1 > g) { g = p1; am = 1; }
        if (p2 > g) { g = p2; am = 2; }
        gate_out[s] = g;
        idx_out[s]  = am;
        atomicAdd(&blk_sum[0], p0);
        atomicAdd(&blk_sum[1], p1);
        atomicAdd(&blk_sum[2], p2);
    }
    __syncthreads();
    if (tid < NEXP) atomicAdd(&sum_gates[tid], blk_sum[tid]);
}

// ---------------------------------------------------------------------------
// 2) routing scan: single 1024-thread workgroup. 3 counters packed into 21-bit
//    fields of u64; Hillis-Steele scan; replay assigns slots + builds perm.
__global__ __launch_bounds__(1024) void moe_scan_kernel(
    const int* __restrict__ idx, float* __restrict__ gate_io,
    int* __restrict__ perm, int* __restrict__ counts, int* __restrict__ route_cnt)
{
    __shared__ unsigned long long sc[1024];    // 8 KB
    const int t = threadIdx.x;
    const int base = t * 32;

    unsigned int c0 = 0, c1 = 0, c2 = 0;
    for (int i = 0; i < 32; ++i) {
        const int e = idx[base + i];
        c0 += (e == 0); c1 += (e == 1); c2 += (e == 2);
    }
    const unsigned long long v = (unsigned long long)c0 |
                                 ((unsigned long long)c1 << 21) |
                                 ((unsigned long long)c2 << 42);
    sc[t] = v;
    __syncthreads();
    for (int off = 1; off < 1024; off <<= 1) {
        const unsigned long long add = (t >= off) ? sc[t - off] : 0ULL;
        __syncthreads();
        sc[t] += add;
        __syncthreads();
    }
    const unsigned long long inc = sc[t];
    const unsigned long long exc = inc - v;
    if (t == 1023) {
        const unsigned int t0 = (unsigned int)(inc & 0x1FFFFF);
        const unsigned int t1 = (unsigned int)((inc >> 21) & 0x1FFFFF);
        const unsigned int t2 = (unsigned int)((inc >> 42) & 0x1FFFFF);
        route_cnt[0] = (int)t0; route_cnt[1] = (int)t1; route_cnt[2] = (int)t2;
        counts[0] = (t0 < CAP) ? (int)t0 : CAP;
        counts[1] = (t1 < CAP) ? (int)t1 : CAP;
        counts[2] = (t2 < CAP) ? (int)t2 : CAP;
    }
    int o0 = (int)(exc & 0x1FFFFF);
    int o1 = (int)((exc >> 21) & 0x1FFFFF);
    int o2 = (int)((exc >> 42) & 0x1FFFFF);
    for (int i = 0; i < 32; ++i) {
        const int tok = base + i;
        const int e = idx[tok];
        int loc;
        if (e == 0)      loc = o0++;
        else if (e == 1) loc = o1++;
        else             loc = o2++;
        const bool kept = (loc < CAP);
        if (kept) perm[e * CAPR + loc] = tok;
        else      gate_io[tok] = 0.0f;       // dropped -> combine weight 0
    }
}

// ---------------------------------------------------------------------------
// 3) grouped expert GEMM with fp32 WMMA.
//    Block = 8 waves = 64 tokens x 512 features. Wave (mi,ni) computes
//    32 tokens x 128 features = 16 accumulators; each B fragment is reused
//    for 2 WMMAs (8 global b64 loads -> 16 v_wmma per K-step of 4).
//    A tile lives in LDS, pair-interleaved (row r and r+32 adjacent) so the
//    two A fragments of a wave combine into one ds_load_2addr_b64.
__global__ __launch_bounds__(256) void moe_gemm_kernel(
    const float* __restrict__ x, const float* __restrict__ W,
    const float* __restrict__ bias, const int* __restrict__ perm,
    const int* __restrict__ counts, const float* __restrict__ gate,
    float* __restrict__ out)
{
    __shared__ float     As[64 * PITCH];   // ~16.5 KB
    __shared__ long long rowBase[64];      // token row offset (-1 = pad row)
    __shared__ float     rowGate[64];

    const int e  = blockIdx.x / MTILES;
    const int m0 = (blockIdx.x % MTILES) * 64;
    const int count = counts[e];
    if (m0 >= count) return;

    const int tid = threadIdx.x;
    if (tid < 64) {
        const int gm = m0 + tid;
        if (gm < count) {
            const int tok = perm[e * CAPR + gm];
            rowBase[tid] = tok_base(tok);
            rowGate[tid] = gate[tok];
        } else {
            rowBase[tid] = -1;
            rowGate[tid] = 0.0f;
        }
    }
    __syncthreads();

    const int wave = tid >> 5, lane = tid & 31;
    const int mi = wave & 1;            // row subtiles mi*16.. and mi*16+32..
    const int ni = wave >> 1;           // 128-feature group (0..3)
    const int mrow  = lane & 15;
    const int khalf = (lane >> 4) << 1; // lanes 0-15: K {0,1}; 16-31: K {2,3}

    v8f acc[2][8] = {};
    // B fragment source: column n of B == row n of W[e] (h = disp @ W^T)
    const float* Wb = W + (size_t)e * DIM * DIM + (size_t)(ni * 128 + mrow) * DIM;

    for (int k0 = 0; k0 < DIM; k0 += 64) {
        // cooperative gathered A-tile fill: 64 rows x 64 K, float2 granules
        // (pitch 66 floats is 8B-aligned only), pair-interleaved row layout
        for (int f = tid; f < 2048; f += 256) {
            const int r = f >> 5;          // source row 0..63
            const int c = (f & 31) << 1;   // float column
            v2f val = {};
            const long long ib = rowBase[r];
            if (ib >= 0) {
                val = *(const v2f*)(x + ib + k0 + c);
                if (k0 + 64 < DIM && (c & 31) == 0)
                    __builtin_prefetch(x + ib + k0 + 64 + c, 0, 0);
            }
            const int lr = ((r & 31) << 1) | (r >> 5);
            *(v2f*)&As[lr * PITCH + c] = val;
        }
        __syncthreads();

        const float* arow = &As[(mi * 16 + mrow) * 2 * PITCH + khalf];
#pragma unroll
        for (int kk = 0; kk < 64; kk += 4) {
            v2f a0, a1;
            a0.x = arow[kk];
            a0.y = arow[kk + 1];
            a1.x = arow[PITCH + kk];
            a1.y = arow[PITCH + kk + 1];
#pragma unroll
            for (int nt = 0; nt < 8; ++nt) {
                const v2f b = *(const v2f*)(Wb + (size_t)nt * 16 * DIM + k0 + kk + khalf);
                acc[0][nt] = __builtin_amdgcn_wmma_f32_16x16x4_f32(
                    false, a0, false, b, (short)0, acc[0][nt], false, false);
                acc[1][nt] = __builtin_amdgcn_wmma_f32_16x16x4_f32(
                    false, a1, false, b, (short)0, acc[1][nt], false, false);
            }
        }
        __syncthreads();
    }

    // combine + store: C/D layout -> lane = n + 16*(m>=8), vgpr v = m&7
    const int nn = lane & 15;
    const int mhalf = (lane >> 4) << 3;
    const float* bias_e = bias + e * DIM;
#pragma unroll
    for (int nt = 0; nt < 8; ++nt) {
        const int f = ni * 128 + nt * 16 + nn;
        const float bv = bias_e[f];
#pragma unroll
        for (int p = 0; p < 2; ++p) {
#pragma unroll
            for (int v = 0; v < 8; ++v) {
                const int r = mi * 16 + p * 32 + mhalf + v;
                const long long ob = rowBase[r];
                if (ob >= 0) out[ob + f] = rowGate[r] * (acc[p][nt][v] + bv);
            }
        }
    }
}

// ---------------------------------------------------------------------------
// 4) zero dropped rows (gate==0 <=> dropped; kept gates >= 1/3) + l_aux
__global__ __launch_bounds__(256) void moe_finalize_kernel(
    const float* __restrict__ gate, float* __restrict__ out,
    const float* __restrict__ sum_gates, const int* __restrict__ route_cnt)
{
    const int wave = threadIdx.x >> 5, lane = threadIdx.x & 31;
    const int s = blockIdx.x * 8 + wave;
    if (s < S_TOK && gate[s] == 0.0f) {
        float* op = out + tok_base(s);
        const v4f z = {};
#pragma unroll
        for (int i = 0; i < 4; ++i) *(v4f*)(op + i * 128 + lane * 4) = z;
    }
    if (blockIdx.x == 0 && threadIdx.x == 0) {
        float la = 0.0f;
        for (int e = 0; e < NEXP; ++e)
            la += (sum_gates[e] / (float)S_TOK) * ((float)route_cnt[e] / (float)S_TOK);
        out[OUT_ELTS] = la * (float)NEXP;
    }
}

// ---------------------------------------------------------------------------
extern "C" void kernel_launch(void* const* d_in, const int* in_sizes, int n_in,
                              void* d_out, int out_size, void* d_ws, size_t ws_size,
                              hipStream_t stream) {
    const float* x    = (const float*)d_in[0];   // [B,L,N,D]
    const float* wg   = (const float*)d_in[1];   // [D,E]
    const float* W    = (const float*)d_in[2];   // [E,D,D]
    const float* bias = (const float*)d_in[3];   // [E,D]
    float* out = (float*)d_out;

    // workspace layout (elements)
    float* gate_ws   = (float*)d_ws;                       // [S]
    int*   idx_ws    = (int*)d_ws + S_TOK;                 // [S]
    int*   perm_ws   = (int*)d_ws + 2 * S_TOK;             // [E*CAPR]
    int*   counts_ws = perm_ws + NEXP * CAPR;              // [E]
    int*   route_ws  = counts_ws + NEXP;                   // [E]
    float* sumg_ws   = (float*)(route_ws + NEXP);          // [E]

    moe_init_kernel<<<1, 64, 0, stream>>>(sumg_ws);
    moe_gate_kernel<<<S_TOK / 8, 256, 0, stream>>>(x, wg, gate_ws, idx_ws, sumg_ws);
    moe_scan_kernel<<<1, 1024, 0, stream>>>(idx_ws, gate_ws, perm_ws, counts_ws, route_ws);
    moe_gemm_kernel<<<NEXP * MTILES, 256, 0, stream>>>(x, W, bias, perm_ws, counts_ws,
                                                       gate_ws, out);
    moe_finalize_kernel<<<S_TOK / 8, 256, 0, stream>>>(gate_ws, out, sumg_ws, route_ws);
}